// GNN_18665927868956
// MI455X (gfx1250) — compile-verified
//
#include <hip/hip_runtime.h>
#include <hip/hip_bf16.h>

// ---------------------------------------------------------------------------
// GNN (3x RGCN-style conv + mean pool) for MI455X / gfx1250, wave32.
//
// Roofline: node tables are L2-resident (<<192MB); cost is dominated by edge
// gather + f32 atomic scatter. Strategy:
//   L0 (38->64): aggregate-first (38 atomic dims/edge, not 64), then ONE fused
//                WMMA kernel: z1 = agg38@Wrel0^T + x@Wroot0^T + b0.
//   L1 (64->32): transform-first. A-fragment = tanh(z1) computed on load.
//                msg1 = A@Wrel1^T, root1 = A@Wroot1^T. Scatter msg1 (32 dims).
//   L2 (32->16): A-fragment = tanh(root1 + agg1 + b1) on load.
//                msg2/root2, scatter msg2 (16 dims).
//   Pool: (root2 + agg2 + b2) atomically segment-summed into [256,16]+counts,
//         then tanh(mean). No intermediate h buffers ever materialized.
// Atomic traffic: (38+32+16) floats/edge = 275M atomics (vs 358M naive).
// ---------------------------------------------------------------------------

typedef float v2f __attribute__((ext_vector_type(2)));
typedef float v8f __attribute__((ext_vector_type(8)));

#define N_NODES  100000
#define N_EDGES  3200000
#define N_GRAPHS 256

__global__ __launch_bounds__(256)
void zero_f32(float* __restrict__ p, int n) {
    int i = blockIdx.x * blockDim.x + threadIdx.x;
    if (i < n) p[i] = 0.0f;
}

// ---------------------------------------------------------------------------
// Raw-feature edge scatter for layer 0: agg[dst] += x[src], DIN=38.
// 19 float2 chunks per edge (rows are 152B -> 8B aligned).
// ---------------------------------------------------------------------------
__global__ __launch_bounds__(256)
void scatter_edges_raw38(const int* __restrict__ src, const int* __restrict__ dst,
                         const float* __restrict__ x, float* __restrict__ agg) {
    constexpr unsigned CH = 19;
    const unsigned tid = blockIdx.x * blockDim.x + threadIdx.x;
    if (tid >= (unsigned)N_EDGES * CH) return;
    const unsigned e = tid / CH;
    const unsigned c = (tid - e * CH) * 2;
    const float2 v = *reinterpret_cast<const float2*>(x + (size_t)src[e] * 38 + c);
    float* a = agg + (size_t)dst[e] * 38 + c;
    atomicAdd(a + 0, v.x);
    atomicAdd(a + 1, v.y);
}

// Transformed-message scatter: agg[dst] += msg[src], 4 dims/thread.
template <int DOUT>
__global__ __launch_bounds__(256)
void scatter_edges(const int* __restrict__ src, const int* __restrict__ dst,
                   const float* __restrict__ msg, float* __restrict__ agg) {
    constexpr unsigned CH = DOUT / 4;
    const unsigned tid = blockIdx.x * blockDim.x + threadIdx.x;
    if (tid >= (unsigned)N_EDGES * CH) return;
    const unsigned e = tid / CH;
    const unsigned c = (tid - e * CH) * 4;
    const float4 v = *reinterpret_cast<const float4*>(msg + (size_t)src[e] * DOUT + c);
    float* a = agg + (size_t)dst[e] * DOUT + c;
    atomicAdd(a + 0, v.x);
    atomicAdd(a + 1, v.y);
    atomicAdd(a + 2, v.z);
    atomicAdd(a + 3, v.w);
}

// ---------------------------------------------------------------------------
// WMMA fragment conventions (V_WMMA_F32_16X16X4_F32, wave32):
//  A (16x4): lanes 0-15 row M=lane K={kk,kk+1}; lanes 16-31 row M=lane-16
//            K={kk+2,kk+3} (v2f per lane).
//  B (4x16): B[k][n]=W[n][k]; same half/K-pair split across lanes.
//  D (16x16): VGPR g -> row m0+g (lanes 0-15) / m0+8+g (lanes 16-31),
//             col n0+(lane&15).
// All guards are wave-uniform -> EXEC all-ones at every WMMA.
// ---------------------------------------------------------------------------

// Layer 0: z = Arel@Wrel^T + Aroot@Wroot^T + bias  (two A-streams, no act).
template <int DIN, int DOUT>
__global__ __launch_bounds__(256)
void transform_dual(const float* __restrict__ Arel,   // agg38 [n, DIN]
                    const float* __restrict__ Aroot,  // x     [n, DIN]
                    const float* __restrict__ Wrel,   // [DOUT, DIN]
                    const float* __restrict__ Wroot,  // [DOUT, DIN]
                    const float* __restrict__ bias,   // [DOUT]
                    float* __restrict__ zout,         // [n, DOUT]
                    int nrows) {
    constexpr int KSTEPS = (DIN + 3) / 4;
    const int lane = threadIdx.x & 31;
    const int wave = (blockIdx.x * blockDim.x + threadIdx.x) >> 5;
    const int m0   = wave * 16;
    if (m0 >= nrows) return;                 // wave-uniform
    const int half = lane >> 4;
    const int l16  = lane & 15;

    const float* arow = Arel  + (size_t)(m0 + l16) * DIN;
    const float* xrow = Aroot + (size_t)(m0 + l16) * DIN;

    // Pre-build both A fragments once (hoisted out of the N-tile loop).
    v2f afr[KSTEPS], afo[KSTEPS];
#pragma unroll
    for (int s = 0; s < KSTEPS; ++s) {
        const int ka = s * 4 + half * 2;
        afr[s].x = (ka     < DIN) ? arow[ka]     : 0.0f;
        afr[s].y = (ka + 1 < DIN) ? arow[ka + 1] : 0.0f;
        afo[s].x = (ka     < DIN) ? xrow[ka]     : 0.0f;
        afo[s].y = (ka + 1 < DIN) ? xrow[ka + 1] : 0.0f;
    }

    for (int n0 = 0; n0 < DOUT; n0 += 16) {
        const float* wr = Wrel  + (size_t)(n0 + l16) * DIN;
        const float* wo = Wroot + (size_t)(n0 + l16) * DIN;
        v8f acc = {};
#pragma unroll
        for (int s = 0; s < KSTEPS; ++s) {
            const int ka = s * 4 + half * 2;
            v2f br, bo;
            br.x = (ka     < DIN) ? wr[ka]     : 0.0f;
            br.y = (ka + 1 < DIN) ? wr[ka + 1] : 0.0f;
            bo.x = (ka     < DIN) ? wo[ka]     : 0.0f;
            bo.y = (ka + 1 < DIN) ? wo[ka + 1] : 0.0f;
            acc = __builtin_amdgcn_wmma_f32_16x16x4_f32(
                false, afr[s], false, br, (short)0, acc, false, false);
            acc = __builtin_amdgcn_wmma_f32_16x16x4_f32(
                false, afo[s], false, bo, (short)0, acc, false, false);
        }
        const float bcol = bias[n0 + l16];
#pragma unroll
        for (int g = 0; g < 8; ++g) {
            const size_t row = (size_t)(m0 + g + half * 8);
            zout[row * DOUT + n0 + l16] = acc[g] + bcol;
        }
    }
}

// Layers 1/2: A = tanh(Z [+ AggP + biasP]) built on load (activation fused);
// emits msg = A@Wrel^T and root = A@Wroot^T.
template <int DIN, int DOUT, bool ADD_AGG>
__global__ __launch_bounds__(256)
void transform_actin(const float* __restrict__ Z,      // [n, DIN] pre-act root part
                     const float* __restrict__ AggP,   // [n, DIN] or unused
                     const float* __restrict__ biasP,  // [DIN] or unused
                     const float* __restrict__ Wrel,   // [DOUT, DIN]
                     const float* __restrict__ Wroot,  // [DOUT, DIN]
                     float* __restrict__ msg,          // [n, DOUT]
                     float* __restrict__ root,         // [n, DOUT]
                     int nrows) {
    constexpr int KSTEPS = (DIN + 3) / 4;
    const int lane = threadIdx.x & 31;
    const int wave = (blockIdx.x * blockDim.x + threadIdx.x) >> 5;
    const int m0   = wave * 16;
    if (m0 >= nrows) return;                 // wave-uniform
    const int half = lane >> 4;
    const int l16  = lane & 15;

    const float* zrow = Z + (size_t)(m0 + l16) * DIN;
    const float* grow = AggP + (size_t)(m0 + l16) * DIN;

    // A fragment with fused activation, built once per wave.
    v2f af[KSTEPS];
#pragma unroll
    for (int s = 0; s < KSTEPS; ++s) {
        const int ka = s * 4 + half * 2;
        float a0 = 0.0f, a1 = 0.0f;
        if (ka < DIN) {
            a0 = zrow[ka];
            if (ADD_AGG) a0 += grow[ka] + biasP[ka];
            a0 = tanhf(a0);
        }
        if (ka + 1 < DIN) {
            a1 = zrow[ka + 1];
            if (ADD_AGG) a1 += grow[ka + 1] + biasP[ka + 1];
            a1 = tanhf(a1);
        }
        af[s].x = a0;
        af[s].y = a1;
    }

    for (int n0 = 0; n0 < DOUT; n0 += 16) {
        const float* wr = Wrel  + (size_t)(n0 + l16) * DIN;
        const float* wo = Wroot + (size_t)(n0 + l16) * DIN;
        v8f acc_rel  = {};
        v8f acc_root = {};
#pragma unroll
        for (int s = 0; s < KSTEPS; ++s) {
            const int ka = s * 4 + half * 2;
            v2f br, bo;
            br.x = (ka     < DIN) ? wr[ka]     : 0.0f;
            br.y = (ka + 1 < DIN) ? wr[ka + 1] : 0.0f;
            bo.x = (ka     < DIN) ? wo[ka]     : 0.0f;
            bo.y = (ka + 1 < DIN) ? wo[ka + 1] : 0.0f;
            acc_rel = __builtin_amdgcn_wmma_f32_16x16x4_f32(
                false, af[s], false, br, (short)0, acc_rel, false, false);
            acc_root = __builtin_amdgcn_wmma_f32_16x16x4_f32(
                false, af[s], false, bo, (short)0, acc_root, false, false);
        }
#pragma unroll
        for (int g = 0; g < 8; ++g) {
            const size_t row = (size_t)(m0 + g + half * 8);
            msg [row * DOUT + n0 + l16] = acc_rel[g];
            root[row * DOUT + n0 + l16] = acc_root[g];
        }
    }
}

// Pool: val = root2 + agg2 + b2 (final layer has no activation), atomic
// segment-sum into [256,16] plus per-graph counts.
__global__ __launch_bounds__(256)
void pool_accum_fused(const float* __restrict__ root2, const float* __restrict__ agg2,
                      const float* __restrict__ b2, const int* __restrict__ batch,
                      float* __restrict__ pool, float* __restrict__ cnt) {
    int i = blockIdx.x * blockDim.x + threadIdx.x;
    if (i >= N_NODES * 16) return;
    const int n = i >> 4, d = i & 15;
    const int b = batch[n];
    atomicAdd(&pool[b * 16 + d], root2[i] + agg2[i] + b2[d]);
    if (d == 0) atomicAdd(&cnt[b], 1.0f);
}

__global__ __launch_bounds__(256)
void pool_finish(const float* __restrict__ pool, const float* __restrict__ cnt,
                 float* __restrict__ out) {
    int i = blockIdx.x * blockDim.x + threadIdx.x;
    if (i >= N_GRAPHS * 16) return;
    const float c = fmaxf(cnt[i >> 4], 1.0f);
    out[i] = tanhf(pool[i] / c);
}

static inline int cdiv_ll(long long a, long long b) { return (int)((a + b - 1) / b); }

extern "C" void kernel_launch(void* const* d_in, const int* in_sizes, int n_in,
                              void* d_out, int out_size, void* d_ws, size_t ws_size,
                              hipStream_t stream) {
    const float* x      = (const float*)d_in[0];
    const int*   edge   = (const int*)d_in[1];      // [2, E]
    const int*   batch  = (const int*)d_in[2];
    const float* Wrel0  = (const float*)d_in[3];
    const float* b0     = (const float*)d_in[4];
    const float* Wroot0 = (const float*)d_in[5];
    const float* Wrel1  = (const float*)d_in[6];
    const float* b1     = (const float*)d_in[7];
    const float* Wroot1 = (const float*)d_in[8];
    const float* Wrel2  = (const float*)d_in[9];
    const float* b2     = (const float*)d_in[10];
    const float* Wroot2 = (const float*)d_in[11];

    const int* src = edge;
    const int* dst = edge + N_EDGES;

    // Workspace (floats): agg(Nx64, reused 38/32/16) | z1(Nx64) | z2(Nx32)
    //                   | z3(Nx16) | msg(Nx32, reused 32/16) | pool | cnt
    float* ws   = (float*)d_ws;
    float* agg  = ws;                               // N x 64 (max reuse)
    float* z1   = agg + (size_t)N_NODES * 64;       // N x 64
    float* z2   = z1  + (size_t)N_NODES * 64;       // N x 32
    float* z3   = z2  + (size_t)N_NODES * 32;       // N x 16
    float* msg  = z3  + (size_t)N_NODES * 16;       // N x 32
    float* pool = msg + (size_t)N_NODES * 32;       // 256 x 16
    float* cnt  = pool + N_GRAPHS * 16;             // 256

    const int BT    = 256;
    const int waves = N_NODES / 16;                 // 6250 exact M-tiles
    const int tfblk = cdiv_ll((long long)waves * 32, BT);

    // ---- Layer 0: aggregate-first in 38 dims, fused dual-A WMMA transform.
    zero_f32<<<cdiv_ll((long long)N_NODES * 38, BT), BT, 0, stream>>>(agg, N_NODES * 38);
    scatter_edges_raw38<<<cdiv_ll((long long)N_EDGES * 19, BT), BT, 0, stream>>>(
        src, dst, x, agg);
    transform_dual<38, 64><<<tfblk, BT, 0, stream>>>(
        agg, x, Wrel0, Wroot0, b0, z1, N_NODES);    // z1 = pre-activation

    // ---- Layer 1: A = tanh(z1) fused into transform; scatter 32 dims.
    transform_actin<64, 32, false><<<tfblk, BT, 0, stream>>>(
        z1, (const float*)nullptr, (const float*)nullptr,
        Wrel1, Wroot1, msg, z2, N_NODES);           // z2 = root1
    zero_f32<<<cdiv_ll((long long)N_NODES * 32, BT), BT, 0, stream>>>(agg, N_NODES * 32);
    scatter_edges<32><<<cdiv_ll((long long)N_EDGES * 8, BT), BT, 0, stream>>>(
        src, dst, msg, agg);                        // agg = agg1

    // ---- Layer 2: A = tanh(z2 + agg1 + b1) fused; scatter 16 dims.
    transform_actin<32, 16, true><<<tfblk, BT, 0, stream>>>(
        z2, agg, b1, Wrel2, Wroot2, msg, z3, N_NODES);  // z3 = root2
    zero_f32<<<cdiv_ll((long long)N_NODES * 16, BT), BT, 0, stream>>>(agg, N_NODES * 16);
    scatter_edges<16><<<cdiv_ll((long long)N_EDGES * 4, BT), BT, 0, stream>>>(
        src, dst, msg, agg);                        // agg = agg2

    // ---- Global mean pool + tanh (root2 + agg2 + b2 fused into accumulate).
    zero_f32<<<cdiv_ll(N_GRAPHS * 16 + N_GRAPHS, BT), BT, 0, stream>>>(
        pool, N_GRAPHS * 16 + N_GRAPHS);
    pool_accum_fused<<<cdiv_ll((long long)N_NODES * 16, BT), BT, 0, stream>>>(
        z3, agg, b2, batch, pool, cnt);
    pool_finish<<<cdiv_ll(N_GRAPHS * 16, BT), BT, 0, stream>>>(pool, cnt, (float*)d_out);
}